// RNN_7078106104552
// MI455X (gfx1250) — compile-verified
//
#include <hip/hip_runtime.h>

// ---------------------------------------------------------------------------
// Dims (match reference)
// ---------------------------------------------------------------------------
#define SEQ    128
#define BATCH  32
#define NHID   512
#define K2     1024        // 2*NHID (concat input)
#define G4     2048        // 4*NHID (gates)
#define NVOC   32000
#define BH     (BATCH * NHID)      // 16384
#define NWG_LSTM 64

typedef __attribute__((ext_vector_type(2))) float v2f;
typedef __attribute__((ext_vector_type(8))) float v8f;

// f32 WMMA: D(16x16,f32) = A(16x4,f32) x B(4x16,f32) + C
// A layout: lane L (m = L&15, half = L>>4): a[0]=A[m][k+2*half], a[1]=A[m][k+2*half+1]
// B layout: lane L (n = L&15):              b[0]=B[k+2*half][n], b[1]=B[k+2*half+1][n]
// C layout: vgpr r, lane L: C[r + 8*(L>>4)][L&15]
static __device__ __forceinline__ v8f wmma_f32(v2f a, v2f b, v8f c) {
  return __builtin_amdgcn_wmma_f32_16x16x4_f32(false, a, false, b, (short)0, c,
                                               false, false);
}

static __device__ __forceinline__ v8f v8f_zero() {
  v8f z;
#pragma unroll
  for (int i = 0; i < 8; ++i) z[i] = 0.0f;
  return z;
}

// CDNA5 async copy: global -> LDS, 16B per lane, tracked by ASYNCcnt.
// Generic LDS pointer's low 32 bits are the LDS byte offset (aperture layout).
static __device__ __forceinline__ void async_copy_b128(float* lds_dst,
                                                       const float4* gsrc) {
  unsigned lds_off = (unsigned)(unsigned long long)lds_dst;
  asm volatile("global_load_async_to_lds_b128 %0, %1, off"
               :
               : "v"(lds_off), "v"(gsrc)
               : "memory");
}

// ---------------------------------------------------------------------------
// Embedding gather + workspace init (barrier counter, h double-buffers)
// grid: 4096 blocks (one per (s,b) token) x 128 threads
// ---------------------------------------------------------------------------
__global__ __launch_bounds__(128) void embed_init_kernel(
    const int* __restrict__ tokens, const float* __restrict__ embW,
    float* __restrict__ emb, float* __restrict__ hbuf, unsigned* __restrict__ cnt) {
  const int bid = blockIdx.x, tid = threadIdx.x;
  if (bid == 0 && tid == 0) *cnt = 0u;
  if (bid < 128) {  // zero hbuf: 4*BH floats = 16384 float4
    reinterpret_cast<float4*>(hbuf)[bid * 128 + tid] = make_float4(0.f, 0.f, 0.f, 0.f);
  }
  const int tok = tokens[bid];
  float4 v = reinterpret_cast<const float4*>(embW)[(size_t)tok * 128 + tid];
  reinterpret_cast<float4*>(emb)[(size_t)bid * 128 + tid] = v;
}

// ---------------------------------------------------------------------------
// 32x32 LDS-tiled transpose with k-pair packing.  in[R][C] (R = n rows,
// C = k cols) ->  out[(k/2)][R][2]: out[((k>>1)*R + r)*2 + (k&1)] = in[r][k].
// A WMMA lane then fetches its (kb,kb+1) B pair as ONE 8-byte load, with
// consecutive lanes (n) at consecutive 8B addresses (fully coalesced).
// block (32,8)
// ---------------------------------------------------------------------------
__global__ __launch_bounds__(256) void transpose_pack2_kernel(
    const float* __restrict__ in, float* __restrict__ out, int R, int C) {
  __shared__ float tile[32][33];
  const int bx = blockIdx.x * 32;  // col (k) in 'in'
  const int by = blockIdx.y * 32;  // row (n) in 'in'
  const int tx = threadIdx.x, ty = threadIdx.y;
#pragma unroll
  for (int i = 0; i < 32; i += 8)
    tile[ty + i][tx] = in[(size_t)(by + ty + i) * C + (bx + tx)];
  __syncthreads();
#pragma unroll
  for (int i = 0; i < 32; i += 8) {
    int k = bx + ty + i;
    int r = by + tx;
    out[((size_t)(k >> 1) * R + r) * 2 + (k & 1)] = tile[tx][ty + i];
  }
}

// ---------------------------------------------------------------------------
// Persistent LSTM. 64 WGs x 128 threads (4 waves).
// WG = (mtile in {0,1}, 16-hidden-unit slice u0). Wave g computes gate g's
// 16x16 pre-activation tile via f32 WMMA over K=1024 (two 512 chunks staged
// in LDS). Device-wide barrier (atomic counter) between layer phases.
// ---------------------------------------------------------------------------
__global__ __launch_bounds__(128) void lstm_kernel(
    const float* __restrict__ emb, const float* __restrict__ Wt,
    const float* __restrict__ fcb, float* __restrict__ hbuf,
    float* __restrict__ oseq, float* __restrict__ hid_out,
    unsigned* __restrict__ cnt) {
  __shared__ float Xs[16 * 516];        // 16 rows, 512 cols, pad 4 (conflict-free)
  __shared__ float gates[4][16][16];    // pre-activations i,f,o,b
  __shared__ float cst[2][16][16];      // cell state slice, both layers

  const int tid  = threadIdx.x;
  const int lane = tid & 31;
  const int g    = tid >> 5;            // gate index 0..3 (= wave)
  const int m    = lane & 15;
  const int hs   = lane >> 4;           // K-half select
  const int m0   = (blockIdx.x & 1) * 16;        // batch tile base
  const int u0   = (blockIdx.x >> 1) * 16;       // hidden-unit slice base
  const int n    = g * NHID + u0 + m;            // gate-row (B-matrix col)

  float* hb[2] = {hbuf, hbuf + 2 * BH};          // [layer][parity][B][H]

  for (int i = tid; i < 2 * 16 * 16; i += 128) (&cst[0][0][0])[i] = 0.0f;

  unsigned epoch = 0;
  int p = 0;
  for (int t = 0; t < SEQ; ++t) {
    for (int l = 0; l < 2; ++l) {
      const float* inp = (l == 0) ? (emb + (size_t)t * BH) : (hb[0] + (p ^ 1) * BH);
      const float* hpv = hb[l] + p * BH;

      v8f acc = v8f_zero();
#pragma unroll
      for (int half = 0; half < 2; ++half) {
        const float* src = half ? hpv : inp;   // [BATCH][512] rows m0..m0+15
        __syncthreads();
        for (int idx = tid; idx < 16 * 128; idx += 128) {
          int r = idx >> 7, c = idx & 127;
          reinterpret_cast<float4*>(&Xs[r * 516])[c] =
              reinterpret_cast<const float4*>(src)[(m0 + r) * 128 + c];
        }
        __syncthreads();
        // packed pairs: Wl[pair][G4][2]; this half covers pairs [0,256)
        const float* Wl = Wt + (size_t)l * K2 * G4 + (size_t)half * 512 * G4;
#pragma unroll 4
        for (int k = 0; k < 512; k += 4) {
          int kb = k + hs * 2;
          v2f a;
          a.x = Xs[m * 516 + kb];
          a.y = Xs[m * 516 + kb + 1];
          v2f b = *reinterpret_cast<const v2f*>(
              &Wl[((size_t)((k >> 1) + hs) * G4 + n) * 2]);
          acc = wmma_f32(a, b, acc);
        }
      }
      const float bias = fcb[l * G4 + g * NHID + u0 + m];
#pragma unroll
      for (int r = 0; r < 8; ++r) gates[g][r + hs * 8][m] = acc[r] + bias;
      __syncthreads();

      float* hdst = hb[l] + (p ^ 1) * BH;
      for (int e = tid; e < 256; e += 128) {
        int br = e >> 4, u = e & 15;
        float ig = 1.0f / (1.0f + __expf(-gates[0][br][u]));
        float fg = 1.0f / (1.0f + __expf(-gates[1][br][u]));
        float og = 1.0f / (1.0f + __expf(-gates[2][br][u]));
        float bg = tanhf(gates[3][br][u]);
        float cv = fg * cst[l][br][u] + ig * bg;
        float hv = og * tanhf(cv);
        cst[l][br][u] = cv;
        int off = (m0 + br) * NHID + u0 + u;
        hdst[off] = hv;
        if (l == 1) oseq[(size_t)t * BH + off] = hv;
      }

      // ---- device-wide barrier (release h, acquire others' h) ----
      epoch += NWG_LSTM;
      __syncthreads();
      if (tid == 0) {
        __hip_atomic_fetch_add(cnt, 1u, __ATOMIC_ACQ_REL, __HIP_MEMORY_SCOPE_AGENT);
        while (__hip_atomic_load(cnt, __ATOMIC_ACQUIRE, __HIP_MEMORY_SCOPE_AGENT) < epoch)
          __builtin_amdgcn_s_sleep(2);
      }
      __syncthreads();
    }
    p ^= 1;
  }
  // 128 steps -> final h sits in parity 0.  hidden out: [2][BATCH][NHID]
  for (int i = blockIdx.x * 128 + tid; i < 2 * BH; i += NWG_LSTM * 128)
    hid_out[i] = (i < BH) ? hb[0][i] : hb[1][i - BH];
}

// ---------------------------------------------------------------------------
// Decoder GEMM: out[4096][32000] = Y[4096][512] * W^T + bias.
// WG = 128x128 block, 256 threads (8 waves). Wave w owns n-tile w and all
// 8 m-tiles (64 C VGPRs). A staged in double-buffered LDS (K-chunks of 32,
// stride-36 swizzle -> conflict-free b64 reads) via ASYNC global->LDS b128
// copies overlapped with WMMA compute. B: one coalesced b64 per WMMA from
// pair-packed weights (L2-resident). Output (524 MB, streamed once) uses
// non-temporal stores so it does not evict the weight stream from L2.
// ---------------------------------------------------------------------------
__global__ __launch_bounds__(256) void decoder_kernel(
    const float* __restrict__ Y, const float* __restrict__ Wd,
    const float* __restrict__ db, float* __restrict__ out) {
  __shared__ float As[2][128 * 36];
  const int tid  = threadIdx.x;
  const int lane = tid & 31;
  const int w    = tid >> 5;
  const int m    = lane & 15;
  const int hs   = lane >> 4;
  const int m0   = blockIdx.y * 128;
  const int n    = blockIdx.x * 128 + w * 16 + m;

  const float4* Yv = reinterpret_cast<const float4*>(Y);

  v8f acc[8];
#pragma unroll
  for (int i = 0; i < 8; ++i) acc[i] = v8f_zero();

  // prologue: async-stage chunk 0 into buffer 0 (4 b128 per thread)
#pragma unroll
  for (int idx = tid; idx < 128 * 8; idx += 256) {
    int r = idx >> 3, c = idx & 7;
    async_copy_b128(&As[0][r * 36 + c * 4], Yv + (size_t)(m0 + r) * 128 + c);
  }

  for (int ic = 0; ic < 16; ++ic) {
    const int cur = ic & 1;
    if (ic + 1 < 16) {
      // issue next chunk into the other buffer, then wait only for current
#pragma unroll
      for (int idx = tid; idx < 128 * 8; idx += 256) {
        int r = idx >> 3, c = idx & 7;
        async_copy_b128(&As[cur ^ 1][r * 36 + c * 4],
                        Yv + (size_t)(m0 + r) * 128 + (ic + 1) * 8 + c);
      }
      asm volatile("s_wait_asynccnt 0x4" ::: "memory");  // in-order: cur done
    } else {
      asm volatile("s_wait_asynccnt 0x0" ::: "memory");
    }
    __syncthreads();

    const int kc = ic * 32;
#pragma unroll
    for (int k = 0; k < 32; k += 4) {
      int kb = k + hs * 2;
      // packed pairs: Wd[pair][NVOC][2]
      v2f b = *reinterpret_cast<const v2f*>(
          &Wd[((size_t)(((kc + k) >> 1) + hs) * NVOC + n) * 2]);
#pragma unroll
      for (int mt = 0; mt < 8; ++mt) {
        v2f a;
        a.x = As[cur][(mt * 16 + m) * 36 + kb];
        a.y = As[cur][(mt * 16 + m) * 36 + kb + 1];
        acc[mt] = wmma_f32(a, b, acc[mt]);
      }
    }
    __syncthreads();  // all waves done with As[cur] before it is restaged
  }

  const float bias = db[n];
#pragma unroll
  for (int mt = 0; mt < 8; ++mt) {
#pragma unroll
    for (int r = 0; r < 8; ++r) {
      int row = m0 + mt * 16 + r + hs * 8;
      __builtin_nontemporal_store(acc[mt][r] + bias, &out[(size_t)row * NVOC + n]);
    }
  }
}

// ---------------------------------------------------------------------------
// Host launcher
// ---------------------------------------------------------------------------
extern "C" void kernel_launch(void* const* d_in, const int* in_sizes, int n_in,
                              void* d_out, int out_size, void* d_ws, size_t ws_size,
                              hipStream_t stream) {
  (void)in_sizes; (void)n_in; (void)out_size; (void)ws_size;
  const int*   tokens = (const int*)  d_in[0];
  const float* embW   = (const float*)d_in[1];
  const float* fcW    = (const float*)d_in[2];
  const float* fcb    = (const float*)d_in[3];
  const float* decW   = (const float*)d_in[4];
  const float* decb   = (const float*)d_in[5];
  float* out = (float*)d_out;

  // workspace layout
  char* ws = (char*)d_ws;
  size_t off = 0;
  unsigned* cnt = (unsigned*)(ws + off);                 off += 256;
  float* emb  = (float*)(ws + off);  off += (size_t)SEQ * BH * 4;          // 8 MB
  float* Wt   = (float*)(ws + off);  off += (size_t)2 * K2 * G4 * 4;       // 16.8 MB
  float* dWt  = (float*)(ws + off);  off += (size_t)NHID * NVOC * 4;       // 65.5 MB
  float* oseq = (float*)(ws + off);  off += (size_t)SEQ * BH * 4;          // 8 MB
  float* hbuf = (float*)(ws + off);  off += (size_t)4 * BH * 4;            // 256 KB

  embed_init_kernel<<<SEQ * BATCH, 128, 0, stream>>>(tokens, embW, emb, hbuf, cnt);

  dim3 tb(32, 8);
  // fc_W[l]: [2048 n][1024 k] -> Wt[l]: pair-packed [512][2048][2]
  transpose_pack2_kernel<<<dim3(K2 / 32, G4 / 32), tb, 0, stream>>>(
      fcW, Wt, G4, K2);
  transpose_pack2_kernel<<<dim3(K2 / 32, G4 / 32), tb, 0, stream>>>(
      fcW + (size_t)G4 * K2, Wt + (size_t)K2 * G4, G4, K2);
  // dec_W: [32000 n][512 k] -> dWt: pair-packed [256][32000][2]
  transpose_pack2_kernel<<<dim3(NHID / 32, NVOC / 32), tb, 0, stream>>>(
      decW, dWt, NVOC, NHID);

  lstm_kernel<<<NWG_LSTM, 128, 0, stream>>>(
      emb, Wt, fcb, hbuf, oseq, out + (size_t)SEQ * BATCH * NVOC, cnt);

  decoder_kernel<<<dim3(NVOC / 128, (SEQ * BATCH) / 128), 256, 0, stream>>>(
      oseq, dWt, decb, out);
}